// GCN_58411555225973
// MI455X (gfx1250) — compile-verified
//
#include <hip/hip_runtime.h>

typedef float v2f __attribute__((ext_vector_type(2)));
typedef float v8f __attribute__((ext_vector_type(8)));

#define IN_F 128
#define H_F  16
#define N_CL 2

// ---------------- zero workspace ----------------
__global__ void GCN_zero(float* __restrict__ p, long n) {
    long i = (long)blockIdx.x * blockDim.x + threadIdx.x;
    long stride = (long)gridDim.x * blockDim.x;
    for (; i < n; i += stride) p[i] = 0.0f;
}

// ---------------- degree accumulation ----------------
__global__ void GCN_degrees(const int* __restrict__ src, const int* __restrict__ dst,
                            float* __restrict__ degO, float* __restrict__ degI, int nE) {
    int i = blockIdx.x * blockDim.x + threadIdx.x;
    if (i >= nE) return;
    atomicAdd(&degO[src[i]], 1.0f);
    atomicAdd(&degI[dst[i]], 1.0f);
}

// ---------------- deg^-1/2 with clip at 1 ----------------
__global__ void GCN_norms(const float* __restrict__ degO, const float* __restrict__ degI,
                          float* __restrict__ nO, float* __restrict__ nI, int nN) {
    int i = blockIdx.x * blockDim.x + threadIdx.x;
    if (i >= nN) return;
    nO[i] = rsqrtf(fmaxf(degO[i], 1.0f));
    nI[i] = rsqrtf(fmaxf(degI[i], 1.0f));
}

// ---------------- layer-1 dense transform via f32 WMMA ----------------
// h1pre[n,0:16] = (x[n,:] * out_norm[n]) @ W1   , one 16-row tile per wave32,
// K=128 swept in 32 chained V_WMMA_F32_16X16X4_F32 ops.
__global__ __launch_bounds__(256)
void GCN_gemm1_wmma(const float* __restrict__ x, const float* __restrict__ W1,
                    const float* __restrict__ normO, float* __restrict__ h1pre, int nTiles) {
    __shared__ float sW[IN_F * H_F];                       // 8 KB of the 320 KB WGP LDS
    for (int i = threadIdx.x; i < IN_F * H_F; i += blockDim.x) sW[i] = W1[i];
    __builtin_amdgcn_s_barrier();

    const int wave = threadIdx.x >> 5;
    const int lane = threadIdx.x & 31;
    const int tile = blockIdx.x * 8 + wave;
    if (tile >= nTiles) return;

    const int m  = lane & 15;          // A: M row / B,D: N column
    const int kh = (lane >> 4) << 1;   // K sub-offset 0 or 2 per half-wave
    const int row = tile * 16 + m;
    const float s = normO[row];
    const float* __restrict__ xr = x + (long)row * IN_F;

    v8f acc = {};
    #pragma unroll 4
    for (int k0 = 0; k0 < IN_F; k0 += 4) {
        v2f a, b;
        a.x = xr[k0 + kh]     * s;     // A 16x4 tile, f32, fused left-norm
        a.y = xr[k0 + kh + 1] * s;
        b.x = sW[(k0 + kh)     * H_F + m];   // B 4x16 tile (column-major striping)
        b.y = sW[(k0 + kh + 1) * H_F + m];
        acc = __builtin_amdgcn_wmma_f32_16x16x4_f32(
                  false, a, false, b, (short)0, acc, false, false);
    }

    // C/D layout: VGPR r -> M = r (lanes 0-15) / M = 8+r (lanes 16-31), N = lane&15
    const int mhalf = (lane >> 4) * 8;
    float* __restrict__ ob = h1pre + ((long)tile * 16 + mhalf) * H_F + m;
    #pragma unroll
    for (int r = 0; r < 8; ++r) ob[r * H_F] = acc[r];
}

// ---------------- layer-1 SpMM: agg1[dst] += h1pre[src], 4 threads/edge ----------------
__global__ void GCN_spmm1(const int* __restrict__ src, const int* __restrict__ dst,
                          const float* __restrict__ h1pre, float* __restrict__ agg1, int nE) {
    int tid = blockIdx.x * blockDim.x + threadIdx.x;
    int e = tid >> 2;
    if (e >= nE) return;
    int q = (tid & 3) * 4;
    const float4 v = *(const float4*)(h1pre + (long)src[e] * H_F + q);
    float* __restrict__ d = agg1 + (long)dst[e] * H_F + q;
    atomicAdd(d + 0, v.x);
    atomicAdd(d + 1, v.y);
    atomicAdd(d + 2, v.z);
    atomicAdd(d + 3, v.w);
}

// ---------------- relu(agg1*in_norm + b1), fuse out_norm and 16->2 GEMM ----------------
__global__ void GCN_post1_gemm2(const float* __restrict__ agg1, const float* __restrict__ normI,
                                const float* __restrict__ normO, const float* __restrict__ b1,
                                const float* __restrict__ W2, float* __restrict__ h2pre, int nN) {
    int n = blockIdx.x * blockDim.x + threadIdx.x;
    if (n >= nN) return;
    const float ni = normI[n];
    const float no = normO[n];
    const float* __restrict__ a = agg1 + (long)n * H_F;
    float acc0 = 0.0f, acc1 = 0.0f;
    #pragma unroll
    for (int f = 0; f < H_F; ++f) {
        float h = fmaxf(fmaf(a[f], ni, b1[f]), 0.0f) * no;
        acc0 = fmaf(h, W2[f * N_CL + 0], acc0);
        acc1 = fmaf(h, W2[f * N_CL + 1], acc1);
    }
    h2pre[(long)n * N_CL + 0] = acc0;
    h2pre[(long)n * N_CL + 1] = acc1;
}

// ---------------- layer-2 SpMM: 1 thread/edge, 2 floats ----------------
__global__ void GCN_spmm2(const int* __restrict__ src, const int* __restrict__ dst,
                          const float* __restrict__ h2pre, float* __restrict__ agg2, int nE) {
    int e = blockIdx.x * blockDim.x + threadIdx.x;
    if (e >= nE) return;
    const float2 v = *(const float2*)(h2pre + (long)src[e] * N_CL);
    float* __restrict__ d = agg2 + (long)dst[e] * N_CL;
    atomicAdd(d + 0, v.x);
    atomicAdd(d + 1, v.y);
}

// ---------------- per-graph sum + count ----------------
__global__ void GCN_graph_reduce(const float* __restrict__ agg2, const float* __restrict__ normI,
                                 const float* __restrict__ b2, const int* __restrict__ ng,
                                 float* __restrict__ gsum, float* __restrict__ gcnt, int nN) {
    int n = blockIdx.x * blockDim.x + threadIdx.x;
    if (n >= nN) return;
    const float ni = normI[n];
    const int g = ng[n];
    atomicAdd(&gsum[g * N_CL + 0], fmaf(agg2[(long)n * N_CL + 0], ni, b2[0]));
    atomicAdd(&gsum[g * N_CL + 1], fmaf(agg2[(long)n * N_CL + 1], ni, b2[1]));
    atomicAdd(&gcnt[g], 1.0f);
}

__global__ void GCN_finalize(const float* __restrict__ gsum, const float* __restrict__ gcnt,
                             float* __restrict__ out, int nOut) {
    int i = blockIdx.x * blockDim.x + threadIdx.x;
    if (i >= nOut) return;
    out[i] = gsum[i] / fmaxf(gcnt[i >> 1], 1.0f);
}

extern "C" void kernel_launch(void* const* d_in, const int* in_sizes, int n_in,
                              void* d_out, int out_size, void* d_ws, size_t ws_size,
                              hipStream_t stream) {
    const float* x     = (const float*)d_in[0];   // (N,128)
    const float* W1    = (const float*)d_in[1];   // (128,16)
    const float* b1    = (const float*)d_in[2];   // (16,)
    const float* W2    = (const float*)d_in[3];   // (16,2)
    const float* b2    = (const float*)d_in[4];   // (2,)
    const int*   src   = (const int*)d_in[5];     // (E,)
    const int*   dst   = (const int*)d_in[6];     // (E,)
    const int*   ngidx = (const int*)d_in[7];     // (N,)
    float* out = (float*)d_out;

    const int nN = in_sizes[7];
    const int nE = in_sizes[5];

    // carve workspace (floats), ws base assumed >=256B aligned
    float* ws = (float*)d_ws;
    size_t off = 0;
    float* degO  = ws + off; off += (size_t)nN;
    float* degI  = ws + off; off += (size_t)nN;
    float* nO    = ws + off; off += (size_t)nN;
    float* nI    = ws + off; off += (size_t)nN;
    float* h1pre = ws + off; off += (size_t)nN * H_F;
    float* agg1  = ws + off; off += (size_t)nN * H_F;
    float* h2pre = ws + off; off += (size_t)nN * N_CL;
    float* agg2  = ws + off; off += (size_t)nN * N_CL;
    float* gsum  = ws + off; off += 128;
    float* gcnt  = ws + off; off += 64;
    const long totalF = (long)off;

    GCN_zero<<<1024, 256, 0, stream>>>(ws, totalF);

    GCN_degrees<<<(nE + 255) / 256, 256, 0, stream>>>(src, dst, degO, degI, nE);
    GCN_norms<<<(nN + 255) / 256, 256, 0, stream>>>(degO, degI, nO, nI, nN);

    const int nTiles = (nN + 15) / 16;                       // 6250 wave-tiles
    GCN_gemm1_wmma<<<(nTiles + 7) / 8, 256, 0, stream>>>(x, W1, nO, h1pre, nTiles);

    const long t1 = (long)nE * 4;                            // 4 threads/edge
    GCN_spmm1<<<(int)((t1 + 255) / 256), 256, 0, stream>>>(src, dst, h1pre, agg1, nE);

    GCN_post1_gemm2<<<(nN + 255) / 256, 256, 0, stream>>>(agg1, nI, nO, b1, W2, h2pre, nN);

    GCN_spmm2<<<(nE + 255) / 256, 256, 0, stream>>>(src, dst, h2pre, agg2, nE);

    GCN_graph_reduce<<<(nN + 255) / 256, 256, 0, stream>>>(agg2, nI, b2, ngidx, gsum, gcnt, nN);

    GCN_finalize<<<1, 128, 0, stream>>>(gsum, gcnt, out, out_size);
}